// LearnableEMA_60790967108171
// MI455X (gfx1250) — compile-verified
//
#include <hip/hip_runtime.h>
#include <math.h>

// Problem constants from reference setup_inputs(): x[B,T,C] f32, alpha_raw[C] f32.
// Outputs: seasonal[B,T,C] then trend[B,T,C], f32, concatenated in d_out.
#define B_  64
#define T_  720
#define C_  512
#define TCH 16                     // timesteps per LDS chunk
#define NCHUNK (T_ / TCH)          // 45
#define DEPTH 3                    // chunks prefetched ahead of consumption
#define NBUF 4                     // LDS ring buffers per wave (DEPTH + 1)
#define LOADS_PER_CHUNK (TCH / 4)  // 4x b128 async DMA ops per chunk per wave

// CDNA5 ISA 10.2: a flat pointer in the LDS aperture carries the wave-relative
// LDS byte offset in addr[31:0]; async-to-LDS ops take the LDS address in vdst
// relative to the wave's LDS base, i.e. the same space. So truncation gives the DMA target.
__device__ __forceinline__ unsigned lds_off(const void* p) {
  return (unsigned)(unsigned long long)(uintptr_t)p;
}

// Issue one chunk: 16 timesteps x 32 channels x 4B = 2KB, moved by 4
// global_load_async_to_lds_b128 ops (32 lanes x 16B = 512B each).
// Lane (r = lane>>3, g = lane&7) loads row r of each 4-row group, 16B column
// group g. Global rows are C_*4 bytes apart; LDS rows are 128B apart.
__device__ __forceinline__ void async_load_chunk(unsigned long long gbase,
                                                 unsigned lbase, int lane) {
  const int r = lane >> 3;
  const int g = lane & 7;
#pragma unroll
  for (int j = 0; j < LOADS_PER_CHUNK; ++j) {
    unsigned long long ga =
        gbase + (unsigned long long)(unsigned)(((j * 4 + r) * C_ + g * 4) * 4);
    unsigned lo = lbase + (unsigned)((j * 4 + r) * 128 + g * 16);
    asm volatile("global_load_async_to_lds_b128 %0, %1, off"
                 :: "v"(lo), "v"(ga)
                 : "memory");
  }
}

__device__ __forceinline__ void consume_chunk(const float* __restrict__ buf,
                                              int lane, double a, double oma,
                                              double& s,
                                              float* __restrict__ tr,
                                              float* __restrict__ se) {
#pragma unroll
  for (int t = 0; t < TCH; ++t) {
    float  xf = buf[t * 32 + lane];          // ds_load_b32 from staged tile
    double xd = (double)xf;
    s = fma(a, xd, oma * s);                 // f64 recurrence (matches reference)
    float tf = (float)s;
    __builtin_nontemporal_store(tf, tr + (size_t)t * C_);       // write-once
    __builtin_nontemporal_store(xf - tf, se + (size_t)t * C_);  // write-once
  }
}

__global__ __launch_bounds__(64) void LearnableEMA_scan_kernel(
    const float* __restrict__ x, const float* __restrict__ alpha_raw,
    float* __restrict__ seasonal, float* __restrict__ trend) {
  // Per-wave private ring: [wave][buf][t][ch] = 2 * 4 * 16 * 32 * 4B = 16KB/block.
  __shared__ __align__(16) float tile[2][NBUF][TCH][32];

  const int lane = threadIdx.x & 31;
  const int w    = threadIdx.x >> 5;
  const int W    = blockIdx.x * 2 + w;   // wave task id, 0..1023
  const int b    = W >> 4;               // batch row   (64)
  const int c0   = (W & 15) * 32;        // channel group (16 groups of 32)
  const int c    = c0 + lane;

  // alpha = sigmoid(alpha_raw) in f64, matching reference numerics.
  const double a   = 1.0 / (1.0 + exp(-(double)alpha_raw[c]));
  const double oma = 1.0 - a;

  const size_t base_e = (size_t)b * T_ * C_ + (size_t)c0;   // elem off of (b,0,c0)
  const unsigned long long gbase =
      (unsigned long long)(uintptr_t)(x + base_e);

  unsigned lbase[NBUF];
#pragma unroll
  for (int i = 0; i < NBUF; ++i) lbase[i] = lds_off(&tile[w][i][0][0]);

  // Prologue: fill the pipeline DEPTH chunks deep (ASYNCcnt = 4*DEPTH).
#pragma unroll
  for (int p = 0; p < DEPTH; ++p)
    async_load_chunk(gbase + (unsigned long long)p * (TCH * C_ * 4),
                     lbase[p], lane);

  // Seed s = x[b,0,c]; then alpha*x0 + (1-alpha)*x0 == x0, so the loop is
  // branch-free while honoring s_0 = x_0.
  double s = (double)x[base_e + (size_t)lane];

  float* tr = trend    + base_e + (size_t)lane;
  float* se = seasonal + base_e + (size_t)lane;

  for (int k = 0; k < NCHUNK; ++k) {
    const int kn = k + DEPTH;
    if (kn < NCHUNK)
      async_load_chunk(gbase + (unsigned long long)kn * (TCH * C_ * 4),
                       lbase[kn & (NBUF - 1)], lane);

    // Async loads complete in order: waiting ASYNCcnt <= 4*(chunks still in
    // flight beyond k) guarantees chunk k has landed in LDS.
    if (k + 3 < NCHUNK)      asm volatile("s_wait_asynccnt 12" ::: "memory");
    else if (k + 2 < NCHUNK) asm volatile("s_wait_asynccnt 8"  ::: "memory");
    else if (k + 1 < NCHUNK) asm volatile("s_wait_asynccnt 4"  ::: "memory");
    else                     asm volatile("s_wait_asynccnt 0"  ::: "memory");

    consume_chunk(&tile[w][k & (NBUF - 1)][0][0], lane, a, oma, s,
                  tr + (size_t)k * (TCH * C_), se + (size_t)k * (TCH * C_));
    // Buffer reuse is safe without fences: all DS reads of a buffer are
    // consumed (value uses force the dscnt wait) before its next async issue
    // in program order.
  }
}

extern "C" void kernel_launch(void* const* d_in, const int* in_sizes, int n_in,
                              void* d_out, int out_size, void* d_ws, size_t ws_size,
                              hipStream_t stream) {
  (void)in_sizes; (void)n_in; (void)d_ws; (void)ws_size; (void)out_size;
  const float* x         = (const float*)d_in[0];
  const float* alpha_raw = (const float*)d_in[1];
  float* seasonal = (float*)d_out;
  float* trend    = seasonal + (size_t)B_ * T_ * C_;

  // 1024 wave tasks (B * C/32), 2 waves per block -> 512 blocks of 64 threads.
  dim3 grid((B_ * (C_ / 32)) / 2);
  dim3 block(64);
  LearnableEMA_scan_kernel<<<grid, block, 0, stream>>>(x, alpha_raw, seasonal, trend);
}